// LearnedGraph_60971355734090
// MI455X (gfx1250) — compile-verified
//
#include <hip/hip_runtime.h>
#include <math.h>

#define N_ROWS 8192
#define IN_F   512
#define OUT_F  256

typedef float v2f __attribute__((ext_vector_type(2)));
typedef float v8f __attribute__((ext_vector_type(8)));

// ---------------------------------------------------------------------------
// Kernel 1: Sa = sum|w_s|, Sb = sum|w_r|  (deterministic LDS tree), then
//           dinv[i] = (0.5*(a*Sb + b*Sa) - a*b) ^ -1/2   (s==0 -> 1)
// ---------------------------------------------------------------------------
__global__ __launch_bounds__(1024) void k_stats(const float* __restrict__ w_s,
                                                const float* __restrict__ w_r,
                                                float* __restrict__ dinv) {
    __shared__ float sA[1024];
    __shared__ float sB[1024];
    const int t = threadIdx.x;
    float sa = 0.0f, sb = 0.0f;
    for (int i = t; i < N_ROWS; i += 1024) {
        sa += fabsf(w_s[i]);
        sb += fabsf(w_r[i]);
    }
    sA[t] = sa; sB[t] = sb;
    __syncthreads();
    for (int off = 512; off > 0; off >>= 1) {
        if (t < off) { sA[t] += sA[t + off]; sB[t] += sB[t + off]; }
        __syncthreads();
    }
    const float Sa = sA[0], Sb = sB[0];
    for (int i = t; i < N_ROWS; i += 1024) {
        float a = fabsf(w_s[i]);
        float b = fabsf(w_r[i]);
        float s = 0.5f * (a * Sb + b * Sa) - a * b;
        s = (s == 0.0f) ? 1.0f : s;
        dinv[i] = 1.0f / sqrtf(s);
    }
}

// ---------------------------------------------------------------------------
// Kernel 2: H = input @ weight  via V_WMMA_F32_16X16X4_F32 (fp32 matrix core)
// Block = 256 threads = 8 waves; each wave owns one 16x16 tile of H.
// blockIdx.x selects group of 8 M-tiles, blockIdx.y selects the N-tile.
// Weight tile (512x16 fp32 = 32 KB) staged in LDS, shared by all 8 waves.
// ---------------------------------------------------------------------------
__global__ __launch_bounds__(256) void k_gemm(const float* __restrict__ A,
                                              const float* __restrict__ W,
                                              float* __restrict__ H) {
    __shared__ float Bs[IN_F * 16];          // [k][n], n fastest — 32 KB
    const int tid   = threadIdx.x;
    const int nBase = blockIdx.y * 16;

    // Stage weight[:, nBase..nBase+15] into LDS.
    #pragma unroll
    for (int it = 0; it < (IN_F * 16) / 256; ++it) {
        const int idx = tid + it * 256;
        const int k   = idx >> 4;
        const int n   = idx & 15;
        Bs[idx] = W[k * OUT_F + nBase + n];
    }
    __syncthreads();

    const int lane  = tid & 31;
    const int wave  = tid >> 5;
    const int mTile = blockIdx.x * 8 + wave;
    const int nIdx  = lane & 15;             // M index for A-frag, N index for B/C
    const int khalf = (lane >> 4) << 1;      // lanes 16-31 hold K+2,K+3
    const float* aRow = A + (mTile * 16 + nIdx) * IN_F + khalf;

    v8f acc = {};
    #pragma unroll 8
    for (int k = 0; k < IN_F; k += 4) {
        v2f a = *(const v2f*)(aRow + k);     // b64 load, 8B aligned
        v2f b;
        b[0] = Bs[(k + khalf + 0) * 16 + nIdx];
        b[1] = Bs[(k + khalf + 1) * 16 + nIdx];
        acc = __builtin_amdgcn_wmma_f32_16x16x4_f32(
            /*neg_a=*/false, a, /*neg_b=*/false, b,
            /*c_mod=*/(short)0, acc, /*reuse_a=*/false, /*reuse_b=*/false);
    }

    // C/D layout: VGPR j -> row (j + 8*(lane>=16)), col = lane&15.
    const int mBase = mTile * 16 + (lane >> 4) * 8;
    #pragma unroll
    for (int j = 0; j < 8; ++j) {
        H[(mBase + j) * OUT_F + nBase + nIdx] = acc[j];
    }
}

// ---------------------------------------------------------------------------
// Kernel 3: per-block-of-128-rows partial column sums of
//           u = sum_i dinv[i]*|w_r[i]|*H[i,:],  v = sum_i dinv[i]*|w_s[i]|*H[i,:]
// thread t owns column t -> fully coalesced H reads. Deterministic order.
// ---------------------------------------------------------------------------
__global__ __launch_bounds__(256) void k_partial(const float* __restrict__ H,
                                                 const float* __restrict__ w_s,
                                                 const float* __restrict__ w_r,
                                                 const float* __restrict__ dinv,
                                                 float* __restrict__ pu,
                                                 float* __restrict__ pv) {
    const int c  = threadIdx.x;
    const int r0 = blockIdx.x * 128;
    float u = 0.0f, v = 0.0f;
    for (int i = r0; i < r0 + 128; ++i) {
        const float h  = H[i * OUT_F + c];
        const float d  = dinv[i];
        u += d * fabsf(w_r[i]) * h;
        v += d * fabsf(w_s[i]) * h;
    }
    pu[blockIdx.x * OUT_F + c] = u;
    pv[blockIdx.x * OUT_F + c] = v;
}

// ---------------------------------------------------------------------------
// Kernel 4: fold the 64 partials per column (fixed order -> deterministic).
// ---------------------------------------------------------------------------
__global__ __launch_bounds__(256) void k_uv(const float* __restrict__ pu,
                                            const float* __restrict__ pv,
                                            float* __restrict__ u,
                                            float* __restrict__ v) {
    const int c = threadIdx.x;
    float su = 0.0f, sv = 0.0f;
    for (int r = 0; r < 64; ++r) {
        su += pu[r * OUT_F + c];
        sv += pv[r * OUT_F + c];
    }
    u[c] = su; v[c] = sv;
}

// ---------------------------------------------------------------------------
// Kernel 5: in-place epilogue on H (stored in d_out):
//   out[i,c] = dinv[i]*(0.5*(a[i]*u[c] + b[i]*v[c])) - dinv[i]^2*a[i]*b[i]*H[i,c]
// (second term removes the j==i rank-2 contribution == zeroed diagonal)
// ---------------------------------------------------------------------------
__global__ __launch_bounds__(256) void k_final(const float* __restrict__ w_s,
                                               const float* __restrict__ w_r,
                                               const float* __restrict__ dinv,
                                               const float* __restrict__ u,
                                               const float* __restrict__ v,
                                               float* __restrict__ out) {
    const int idx = blockIdx.x * 256 + threadIdx.x;
    const int i = idx >> 8;       // row
    const int c = idx & 255;      // col
    const float a = fabsf(w_s[i]);
    const float b = fabsf(w_r[i]);
    const float d = dinv[i];
    const float h = out[idx];
    out[idx] = d * (0.5f * (a * u[c] + b * v[c])) - d * d * a * b * h;
}

extern "C" void kernel_launch(void* const* d_in, const int* in_sizes, int n_in,
                              void* d_out, int out_size, void* d_ws, size_t ws_size,
                              hipStream_t stream) {
    (void)in_sizes; (void)n_in; (void)out_size; (void)ws_size;
    const float* input  = (const float*)d_in[0];   // [8192, 512]
    const float* w_s    = (const float*)d_in[1];   // [8192, 1]
    const float* w_r    = (const float*)d_in[2];   // [8192, 1]
    const float* weight = (const float*)d_in[3];   // [512, 256]
    float* out = (float*)d_out;                    // [8192, 256] (holds H, then result)

    // Workspace layout (floats): dinv | u | v | pu | pv   (~162 KB total)
    float* ws   = (float*)d_ws;
    float* dinv = ws;                    // 8192
    float* u    = dinv + N_ROWS;         // 256
    float* v    = u + OUT_F;             // 256
    float* pu   = v + OUT_F;             // 64*256
    float* pv   = pu + 64 * OUT_F;       // 64*256

    k_stats  <<<1,             1024, 0, stream>>>(w_s, w_r, dinv);
    k_gemm   <<<dim3(64, 16),  256,  0, stream>>>(input, weight, out);
    k_partial<<<64,            256,  0, stream>>>(out, w_s, w_r, dinv, pu, pv);
    k_uv     <<<1,             256,  0, stream>>>(pu, pv, u, v);
    k_final  <<<8192,          256,  0, stream>>>(w_s, w_r, dinv, u, v, out);
}